// DepthAggregateModule_41532333753069
// MI455X (gfx1250) — compile-verified
//
#include <hip/hip_runtime.h>

// ---- problem constants (from reference setup_inputs) ----
constexpr int Lc = 12;            // layers
constexpr int Bc = 2;
constexpr int Tc = 2048;
constexpr int Dc = 2048;          // model dim
constexpr int Cc = 4;             // ways
constexpr int Hc = Cc * Lc;       // 48
constexpr int Nc = Bc * Tc;       // 4096 tokens
constexpr float EPSc = 1e-6f;

typedef __attribute__((ext_vector_type(2))) float v2f;
typedef __attribute__((ext_vector_type(8))) float v8f;

static __device__ __forceinline__ v8f wmma_f32_4(v2f a, v2f b, v8f c) {
  // D = A(16x4) * B(4x16) + C(16x16), fp32 matrix core
  return __builtin_amdgcn_wmma_f32_16x16x4_f32(
      /*neg_a=*/false, a, /*neg_b=*/false, b,
      /*c_mod=*/(short)0, c, /*reuse_a=*/false, /*reuse_b=*/false);
}

static __device__ __forceinline__ float gelu_exact(float x) {
  return 0.5f * x * (1.0f + erff(x * 0.70710678118654752f));
}

// -------------------------------------------------------------------------
// Kernel 1: dw[token][48] = gelu(rms(x_last) @ w1) @ w2 + static_weight
// One wave (32 threads) per 16-token tile. Both matmuls on v_wmma_f32_16x16x4.
// -------------------------------------------------------------------------
__global__ __launch_bounds__(32) void dw_kernel(
    const float* __restrict__ xs, const float* __restrict__ w1,
    const float* __restrict__ w2, const float* __restrict__ sw,
    float* __restrict__ dw_out) {
  __shared__ float gmat[16 * 49];  // 16 rows x 48 cols, stride 49 (bank-safe)

  const int lane = threadIdx.x;
  const int lo = lane & 15;   // row (A) or col (B/C/D) within 16
  const int hi = lane >> 4;   // 0: K/M-low half, 1: high half
  const long m0 = (long)blockIdx.x * 16;

  const float* xcur = xs + (long)(Lc - 1) * Nc * Dc;
  const float* arow = xcur + (m0 + lo) * (long)Dc + 2 * hi;

  v8f acc[3] = {};
  float ssq = 0.0f;

  // ---- stage 1: S = x_last @ w1, K = 2048 ----
  for (int k = 0; k < Dc; k += 4) {
    v2f a;
    a.x = arow[k];
    a.y = arow[k + 1];
    ssq += a.x * a.x + a.y * a.y;  // lanes l and l^16 together cover a row

    const float* bp = w1 + (long)(k + 2 * hi) * Hc + lo;
#pragma unroll
    for (int j = 0; j < 3; ++j) {
      v2f b;
      b.x = bp[16 * j];       // row k+2*hi
      b.y = bp[16 * j + Hc];  // row k+2*hi+1
      acc[j] = wmma_f32_4(a, b, acc[j]);
    }
  }

  // per-row rms factor: lane (x&15) holds full-row sum after xor-16 add
  float tot = ssq + __shfl_xor(ssq, 16, 32);
  float rinv = rsqrtf(tot * (1.0f / (float)Dc) + EPSc);

  // ---- gelu(rinv * S) -> LDS ----
#pragma unroll
  for (int j = 0; j < 3; ++j) {
#pragma unroll
    for (int r = 0; r < 8; ++r) {
      int row = r + 8 * hi;
      float ri = __shfl(rinv, row, 32);
      float s = acc[j][r] * ri;
      gmat[row * 49 + j * 16 + lo] = gelu_exact(s);
    }
  }
  __syncthreads();

  // ---- stage 2: dw = G(16x48) @ w2(48x48), K = 48 ----
  v8f dcc[3] = {};
  for (int kk = 0; kk < Hc; kk += 4) {
    v2f a;
    a.x = gmat[lo * 49 + kk + 2 * hi];
    a.y = gmat[lo * 49 + kk + 2 * hi + 1];
    const float* bp = w2 + (long)(kk + 2 * hi) * Hc + lo;
#pragma unroll
    for (int j = 0; j < 3; ++j) {
      v2f b;
      b.x = bp[16 * j];
      b.y = bp[16 * j + Hc];
      dcc[j] = wmma_f32_4(a, b, dcc[j]);
    }
  }

  // ---- + static_weight (flat [C,L] == col index), store [token][48] ----
#pragma unroll
  for (int j = 0; j < 3; ++j) {
    int col = j * 16 + lo;
    float st = sw[col];
#pragma unroll
    for (int r = 0; r < 8; ++r) {
      int row = r + 8 * hi;
      dw_out[(m0 + row) * Hc + col] = dcc[j][r] + st;
    }
  }
}

// -------------------------------------------------------------------------
// Kernel 2: per-token streaming aggregate.
// One 256-thread workgroup per token; the 12x2048 fp32 tile (96KB) is pulled
// into LDS with GLOBAL_LOAD_ASYNC_TO_LDS_B128 (ASYNCcnt, no VGPR staging),
// issued as one burst, then a single s_wait_asynccnt + barrier.
// NOTE: this kernel has no static __shared__, so dynamic LDS starts at byte
// offset 0 and the raw offsets used by the async copies match smem[].
// -------------------------------------------------------------------------
static __device__ __forceinline__ float wave_red(float v) {
#pragma unroll
  for (int o = 16; o > 0; o >>= 1) v += __shfl_xor(v, o, 32);
  return v;
}

__global__ __launch_bounds__(256) void agg_kernel(
    const float* __restrict__ xs, const float* __restrict__ pre,
    const float* __restrict__ post, const float* __restrict__ dw_in,
    float* __restrict__ out) {
  extern __shared__ float smem[];
  float* xt = smem;                   // 12*2048 floats, LDS byte offset 0
  float* wred = xt + Lc * Dc;         // 12*8
  float* cred = wred + Lc * 8;        // 4*8
  float* rinv = cred + Cc * 8;        // 12
  float* rpost = rinv + Lc;           // 4
  float* dwt = rpost + Cc;            // 48

  const int tid = threadIdx.x;
  const int w = tid >> 5;
  const int lane = tid & 31;
  const long n = blockIdx.x;  // token index (b*T + t)

  if (tid < Hc) dwt[tid] = dw_in[n * Hc + tid];

  float4* xt4 = (float4*)xt;
  const int D4 = Dc / 4;  // 512

  // ---- async burst: 12 rows x 8KB, global -> LDS, b128 per thread-slot ----
#pragma unroll
  for (int l = 0; l < Lc; ++l) {
#pragma unroll
    for (int v = 0; v < 2; ++v) {
      int c4 = v * 256 + tid;
      unsigned lds_off = (unsigned)((l * D4 + c4) * 16);  // byte addr in LDS
      unsigned long long ga =
          (unsigned long long)(const void*)(xs + ((long)l * Nc + n) * Dc +
                                            (long)c4 * 4);
      asm volatile("global_load_async_to_lds_b128 %0, %1, off"
                   :
                   : "v"(lds_off), "v"(ga)
                   : "memory");
    }
  }
  asm volatile("s_wait_asynccnt 0x0" ::: "memory");
  __syncthreads();

  // ---- per-layer sum of squares from LDS ----
#pragma unroll 1
  for (int l = 0; l < Lc; ++l) {
    float loc = 0.0f;
#pragma unroll
    for (int v = 0; v < 2; ++v) {
      float4 x = xt4[l * D4 + v * 256 + tid];
      loc += x.x * x.x + x.y * x.y + x.z * x.z + x.w * x.w;
    }
    float s = wave_red(loc);
    if (lane == 0) wred[l * 8 + w] = s;
  }
  __syncthreads();
  if (tid < Lc) {
    float s = 0.0f;
#pragma unroll
    for (int i = 0; i < 8; ++i) s += wred[tid * 8 + i];
    rinv[tid] = rsqrtf(s * (1.0f / (float)Dc) + EPSc);
  }
  __syncthreads();

  // ---- agg[c][d] = sum_l dw[c,l] * rinv_l * pre[l,d] * x[l,d] ----
  const float4* pre4 = (const float4*)pre;
  float4 acc[2][4];
#pragma unroll
  for (int v = 0; v < 2; ++v)
#pragma unroll
    for (int c = 0; c < Cc; ++c) acc[v][c] = make_float4(0.f, 0.f, 0.f, 0.f);

#pragma unroll
  for (int v = 0; v < 2; ++v) {
    int c4 = v * 256 + tid;
#pragma unroll 1
    for (int l = 0; l < Lc; ++l) {
      float4 x = xt4[l * D4 + c4];
      float4 p = pre4[l * D4 + c4];
      float r = rinv[l];
      float xnx = x.x * p.x * r;
      float xny = x.y * p.y * r;
      float xnz = x.z * p.z * r;
      float xnw = x.w * p.w * r;
#pragma unroll
      for (int c = 0; c < Cc; ++c) {
        float wv = dwt[c * Lc + l];
        acc[v][c].x += wv * xnx;
        acc[v][c].y += wv * xny;
        acc[v][c].z += wv * xnz;
        acc[v][c].w += wv * xnw;
      }
    }
  }

  // ---- post rms per way ----
#pragma unroll
  for (int c = 0; c < Cc; ++c) {
    float s = 0.0f;
#pragma unroll
    for (int v = 0; v < 2; ++v)
      s += acc[v][c].x * acc[v][c].x + acc[v][c].y * acc[v][c].y +
           acc[v][c].z * acc[v][c].z + acc[v][c].w * acc[v][c].w;
    s = wave_red(s);
    if (lane == 0) cred[c * 8 + w] = s;
  }
  __syncthreads();
  if (tid < Cc) {
    float s = 0.0f;
#pragma unroll
    for (int i = 0; i < 8; ++i) s += cred[tid * 8 + i];
    rpost[tid] = rsqrtf(s * (1.0f / (float)Dc) + EPSc);
  }
  __syncthreads();

  // ---- out = rms(agg)*post_scales + x_last ----
  const float4* post4 = (const float4*)post;
  float4* out4 = (float4*)out;
#pragma unroll
  for (int v = 0; v < 2; ++v) {
    int c4 = v * 256 + tid;
    float4 xc = xt4[(Lc - 1) * D4 + c4];
#pragma unroll
    for (int c = 0; c < Cc; ++c) {
      float rp = rpost[c];
      float4 po = post4[c * D4 + c4];
      float4 o;
      o.x = acc[v][c].x * rp * po.x + xc.x;
      o.y = acc[v][c].y * rp * po.y + xc.y;
      o.z = acc[v][c].z * rp * po.z + xc.z;
      o.w = acc[v][c].w * rp * po.w + xc.w;
      out4[((long)c * Nc + n) * D4 + c4] = o;
    }
  }
}

// -------------------------------------------------------------------------
extern "C" void kernel_launch(void* const* d_in, const int* in_sizes, int n_in,
                              void* d_out, int out_size, void* d_ws,
                              size_t ws_size, hipStream_t stream) {
  const float* xs = (const float*)d_in[0];
  const float* w1 = (const float*)d_in[1];
  const float* w2 = (const float*)d_in[2];
  const float* sw = (const float*)d_in[3];
  const float* pre = (const float*)d_in[4];
  const float* post = (const float*)d_in[5];
  float* out = (float*)d_out;
  float* dw_ws = (float*)d_ws;  // Nc * 48 floats = 768 KB

  dw_kernel<<<Nc / 16, 32, 0, stream>>>(xs, w1, w2, sw, dw_ws);

  constexpr size_t smem_bytes =
      (size_t)(Lc * Dc + Lc * 8 + Cc * 8 + Lc + Cc + Hc) * sizeof(float);
  agg_kernel<<<Nc, 256, smem_bytes, stream>>>(xs, pre, post, dw_ws, out);
}